// AtomLevelAttention_75299366633812
// MI455X (gfx1250) — compile-verified
//
#include <hip/hip_runtime.h>
#include <hip/hip_bf16.h>
#include <stdint.h>

typedef __attribute__((ext_vector_type(2))) float v2f;
typedef __attribute__((ext_vector_type(8))) float v8f;

#define N_NODES  500000
#define N_GRAPHS 16384
#define EMB      128
#define TOPK     16
#define LDSB_STRIDE 132   // transposed B: Bt[col*132 + k]; conflict-free b64 reads

// ---------------------------------------------------------------------------
// K0: tiny prep — fold proto/b1 into cb[128], collapse W2@w_score into v[128],
//     scalar c = b2.w_score + b_score.
// ---------------------------------------------------------------------------
__global__ __launch_bounds__(128) void prep_kernel(
    const float* __restrict__ prot, const float* __restrict__ W1,
    const float* __restrict__ b1,   const float* __restrict__ W2,
    const float* __restrict__ b2,   const float* __restrict__ w_score,
    const float* __restrict__ b_score,
    float* __restrict__ cb, float* __restrict__ v, float* __restrict__ cconst) {
  const int d = threadIdx.x;  // 0..127
  float s = b1[d];
  for (int j = 0; j < EMB; ++j)
    s += 0.5f * (prot[j] + prot[EMB + j]) * W1[(2 * EMB + j) * EMB + d];
  cb[d] = s;
  float sv = 0.f;
  for (int j = 0; j < EMB; ++j) sv += W2[d * EMB + j] * w_score[j];
  v[d] = sv;
  if (d == 0) {
    float c = b_score[0];
    for (int j = 0; j < EMB; ++j) c += b2[j] * w_score[j];
    *cconst = c;
  }
}

// ---------------------------------------------------------------------------
// Cooperative transposed load of a 128x128 f32 B matrix into padded LDS:
// ldsBt[col * LDSB_STRIDE + k] = B[k][col]. (128 threads)
// ---------------------------------------------------------------------------
__device__ inline void load_B_lds_T(const float* __restrict__ Bsrc, float* ldsBt) {
  const int tid = threadIdx.x;
#pragma unroll
  for (int i = 0; i < 32; ++i) {
    const int idx = (i * 128 + tid) * 4;      // flat float offset
    const int k   = idx >> 7;                 // source row (K index)
    const int col = idx & 127;                // source col base
    const float4 val = *(const float4*)(Bsrc + idx);
    ldsBt[(col + 0) * LDSB_STRIDE + k] = val.x;
    ldsBt[(col + 1) * LDSB_STRIDE + k] = val.y;
    ldsBt[(col + 2) * LDSB_STRIDE + k] = val.z;
    ldsBt[(col + 3) * LDSB_STRIDE + k] = val.w;
  }
}

// ---------------------------------------------------------------------------
// One wave: C[16x128] += A[16x128] @ B[128x128], B transposed in padded LDS.
// V_WMMA_F32_16X16X4_F32, 8 col-tiles x 32 k-steps.
// A fragment (16x4): lane l -> row l&15, K = 4kk + 2*(l>>4) + {0,1} -> global float2
// B fragment (4x16): lane l -> col l&15, same K pair -> single ds_load_b64
// ---------------------------------------------------------------------------
__device__ inline void wave_gemm_16x128(const float* __restrict__ A,
                                        const float* __restrict__ ldsBt,
                                        v8f acc[8]) {
  const int lane = threadIdx.x & 31;
  const int half = lane >> 4;
  const int lm   = lane & 15;
  const float* arow = A + (size_t)lm * EMB;
  const float* bbase[8];
#pragma unroll
  for (int t = 0; t < 8; ++t) bbase[t] = ldsBt + (16 * t + lm) * LDSB_STRIDE;
#pragma unroll 4
  for (int kk = 0; kk < 32; ++kk) {
    const int k0 = 4 * kk + 2 * half;
    const v2f a = *(const v2f*)(arow + k0);       // global 8B load, aligned pair
#pragma unroll
    for (int t = 0; t < 8; ++t) {
      const v2f b = *(const v2f*)(bbase[t] + k0); // ds_load_b64, aligned pair
      acc[t] = __builtin_amdgcn_wmma_f32_16x16x4_f32(
          false, a, false, b, (short)0, acc[t], false, false);
    }
  }
}

// ---------------------------------------------------------------------------
// K1: gc[G,128] = graph_repr @ W1[128:256,:]   (one 16-row tile per wave)
// ---------------------------------------------------------------------------
__global__ __launch_bounds__(128) void graph_gemm_kernel(
    const float* __restrict__ graph_repr, const float* __restrict__ W1mid,
    float* __restrict__ gc) {
  __shared__ float ldsBt[128 * LDSB_STRIDE];
  load_B_lds_T(W1mid, ldsBt);
  __syncthreads();
  const int wave = threadIdx.x >> 5;
  const int row_base = (blockIdx.x * 4 + wave) * 16;
  const int lane = threadIdx.x & 31;
  const int half = lane >> 4;
  const int lm   = lane & 15;
  v8f acc[8];
#pragma unroll
  for (int t = 0; t < 8; ++t)
#pragma unroll
    for (int r = 0; r < 8; ++r) acc[t][r] = 0.f;
  wave_gemm_16x128(graph_repr + (size_t)row_base * EMB, ldsBt, acc);
#pragma unroll
  for (int t = 0; t < 8; ++t)
#pragma unroll
    for (int r = 0; r < 8; ++r)
      gc[(size_t)(row_base + half * 8 + r) * EMB + 16 * t + lm] = acc[t][r];
}

// ---------------------------------------------------------------------------
// K2: score[n] = relu(node@W1top + gc[batch[n]] + cb) . v + c
//     Hidden activations never hit memory.
// ---------------------------------------------------------------------------
__global__ __launch_bounds__(128) void score_kernel(
    const float* __restrict__ node_repr, const float* __restrict__ W1top,
    const int* __restrict__ batch, const float* __restrict__ gc,
    const float* __restrict__ cb, const float* __restrict__ vvec,
    const float* __restrict__ cconst, float* __restrict__ score) {
  __shared__ float ldsBt[128 * LDSB_STRIDE];
  load_B_lds_T(W1top, ldsBt);
  __syncthreads();
  const int wave = threadIdx.x >> 5;
  const int row_base = (blockIdx.x * 4 + wave) * 16;
  if (row_base >= N_NODES) return;
  const int lane = threadIdx.x & 31;
  const int half = lane >> 4;
  const int lm   = lane & 15;

  v8f acc[8];
#pragma unroll
  for (int t = 0; t < 8; ++t)
#pragma unroll
    for (int r = 0; r < 8; ++r) acc[t][r] = 0.f;
  wave_gemm_16x128(node_repr + (size_t)row_base * EMB, ldsBt, acc);

  float cbv[8], vv[8];
#pragma unroll
  for (int t = 0; t < 8; ++t) {
    cbv[t] = cb[16 * t + lm];
    vv[t]  = vvec[16 * t + lm];
  }

  float partial[8];
#pragma unroll
  for (int r = 0; r < 8; ++r) partial[r] = 0.f;

  const int gfirst = batch[row_base];
  const int glast  = batch[row_base + 15];
  if (gfirst == glast) {           // common case: tile entirely in one graph
    const float* gcr = gc + (size_t)gfirst * EMB;
#pragma unroll
    for (int t = 0; t < 8; ++t) {
      const float base = gcr[16 * t + lm] + cbv[t];
#pragma unroll
      for (int r = 0; r < 8; ++r) {
        float h = acc[t][r] + base;
        h = h > 0.f ? h : 0.f;
        partial[r] += h * vv[t];
      }
    }
  } else {
    int grow[8];
#pragma unroll
    for (int r = 0; r < 8; ++r) grow[r] = batch[row_base + half * 8 + r];
#pragma unroll
    for (int t = 0; t < 8; ++t)
#pragma unroll
      for (int r = 0; r < 8; ++r) {
        float h = acc[t][r] + gc[(size_t)grow[r] * EMB + 16 * t + lm] + cbv[t];
        h = h > 0.f ? h : 0.f;
        partial[r] += h * vv[t];
      }
  }
#pragma unroll
  for (int r = 0; r < 8; ++r) {
    float p = partial[r];
    p += __shfl_xor(p, 1, 32);
    p += __shfl_xor(p, 2, 32);
    p += __shfl_xor(p, 4, 32);
    p += __shfl_xor(p, 8, 32);
    partial[r] = p;
  }
  if (lm == 0) {
    const float c = *cconst;
#pragma unroll
    for (int r = 0; r < 8; ++r)
      score[row_base + half * 8 + r] = partial[r] + c;
  }
}

// ---------------------------------------------------------------------------
// K3: segment softmax. One wave per graph; batch is sorted -> binary search.
// ---------------------------------------------------------------------------
__global__ __launch_bounds__(128) void softmax_kernel(
    const int* __restrict__ batch, const float* __restrict__ score,
    float* __restrict__ w_out, int2* __restrict__ ranges) {
  const int g = (blockIdx.x * blockDim.x + threadIdx.x) >> 5;
  const int lane = threadIdx.x & 31;
  if (g >= N_GRAPHS) return;
  int lo = 0, hi = N_NODES;
  while (lo < hi) { int mid = (lo + hi) >> 1; if (batch[mid] < g) lo = mid + 1; else hi = mid; }
  const int start = lo;
  hi = N_NODES;
  while (lo < hi) { int mid = (lo + hi) >> 1; if (batch[mid] < g + 1) lo = mid + 1; else hi = mid; }
  const int end = lo;
  if (lane == 0) { int2 r; r.x = start; r.y = end; ranges[g] = r; }
  float mx = -INFINITY;
  for (int i = start + lane; i < end; i += 32) mx = fmaxf(mx, score[i]);
#pragma unroll
  for (int off = 16; off; off >>= 1) mx = fmaxf(mx, __shfl_xor(mx, off, 32));
  float s = 0.f;
  for (int i = start + lane; i < end; i += 32) s += __expf(score[i] - mx);
#pragma unroll
  for (int off = 16; off; off >>= 1) s += __shfl_xor(s, off, 32);
  const float inv = 1.0f / s;
  for (int i = start + lane; i < end; i += 32) w_out[i] = __expf(score[i] - mx) * inv;
}

// ---------------------------------------------------------------------------
// K4: iterative segment top-k (max-w, min-index tie-break, N-pad) + weighted
//     gather-reduce sub[g] = sum_k sel_w * node_repr[sel]. One wave per graph.
// ---------------------------------------------------------------------------
__global__ __launch_bounds__(128) void topk_kernel(
    const float* __restrict__ node_repr, const float* __restrict__ w,
    const int2* __restrict__ ranges, float* __restrict__ sub,
    float* __restrict__ sel_idx_out, float* __restrict__ sel_w_out) {
  const int g = (blockIdx.x * blockDim.x + threadIdx.x) >> 5;
  const int lane = threadIdx.x & 31;
  if (g >= N_GRAPHS) return;
  const int2 rg = ranges[g];
  int sel[TOPK];
  float selw[TOPK];
#pragma unroll
  for (int k = 0; k < TOPK; ++k) {
    float bw = -INFINITY; int bi = N_NODES;
    for (int i = rg.x + lane; i < rg.y; i += 32) {
      bool taken = false;
#pragma unroll
      for (int j = 0; j < TOPK; ++j)
        if (j < k && sel[j] == i) taken = true;
      if (taken) continue;
      const float wi = w[i];
      if (wi > bw || (wi == bw && i < bi)) { bw = wi; bi = i; }
    }
#pragma unroll
    for (int off = 16; off; off >>= 1) {
      const float ow = __shfl_xor(bw, off, 32);
      const int   oi = __shfl_xor(bi, off, 32);
      if (ow > bw || (ow == bw && oi < bi)) { bw = ow; bi = oi; }
    }
    sel[k]  = bi;
    selw[k] = (bi < N_NODES) ? bw : 0.f;
    if (lane == 0) {
      sel_idx_out[g * TOPK + k] = (float)bi;
      sel_w_out[g * TOPK + k]   = selw[k];
    }
  }
  float4 acc; acc.x = acc.y = acc.z = acc.w = 0.f;
#pragma unroll
  for (int k = 0; k < TOPK; ++k) {
    if (sel[k] < N_NODES) {
      const float4 nv = *(const float4*)(node_repr + (size_t)sel[k] * EMB + lane * 4);
      acc.x += selw[k] * nv.x; acc.y += selw[k] * nv.y;
      acc.z += selw[k] * nv.z; acc.w += selw[k] * nv.w;
    }
  }
  *(float4*)(sub + (size_t)g * EMB + lane * 4) = acc;
}

// ---------------------------------------------------------------------------
extern "C" void kernel_launch(void* const* d_in, const int* in_sizes, int n_in,
                              void* d_out, int out_size, void* d_ws, size_t ws_size,
                              hipStream_t stream) {
  const float* node_repr  = (const float*)d_in[0];
  const float* graph_repr = (const float*)d_in[1];
  const float* prototypes = (const float*)d_in[2];
  const int*   batch      = (const int*)d_in[3];
  const float* W1         = (const float*)d_in[4];
  const float* b1         = (const float*)d_in[5];
  const float* W2         = (const float*)d_in[6];
  const float* b2         = (const float*)d_in[7];
  const float* w_score    = (const float*)d_in[8];
  const float* b_score    = (const float*)d_in[9];

  // workspace layout
  char*  ws     = (char*)d_ws;
  float* gc     = (float*)ws;                                         // G*128
  float* score  = (float*)(ws + (size_t)N_GRAPHS * EMB * 4);          // N
  float* cb     = (float*)(ws + (size_t)N_GRAPHS * EMB * 4 + (size_t)N_NODES * 4);
  float* v      = cb + EMB;
  float* cconst = v + EMB;
  int2*  ranges = (int2*)(((uintptr_t)(cconst + 4) + 15) & ~(uintptr_t)15);

  // output layout: sub[G,128] | w[N] | sel_idx[G,16] | sel_w[G,16]
  float* sub         = (float*)d_out;
  float* w_out       = sub + (size_t)N_GRAPHS * EMB;
  float* sel_idx_out = w_out + N_NODES;
  float* sel_w_out   = sel_idx_out + (size_t)N_GRAPHS * TOPK;

  prep_kernel<<<1, 128, 0, stream>>>(prototypes, W1, b1, W2, b2, w_score,
                                     b_score, cb, v, cconst);
  graph_gemm_kernel<<<N_GRAPHS / 64, 128, 0, stream>>>(graph_repr,
                                                       W1 + EMB * EMB, gc);
  const int ntiles = N_NODES / 16;  // 31250 exact
  score_kernel<<<(ntiles + 3) / 4, 128, 0, stream>>>(node_repr, W1, batch, gc,
                                                     cb, v, cconst, score);
  softmax_kernel<<<N_GRAPHS / 4, 128, 0, stream>>>(batch, score, w_out, ranges);
  topk_kernel<<<N_GRAPHS / 4, 128, 0, stream>>>(node_repr, w_out, ranges, sub,
                                                sel_idx_out, sel_w_out);
}